// RenderMesh_17368847745632
// MI455X (gfx1250) — compile-verified
//
#include <hip/hip_runtime.h>
#include <math.h>

#define IMG   128
#define HW    (IMG*IMG)
#define NB    2
#define NV    1300
#define NF    2500
#define NFP   2512              // faces padded to multiple of 16
#define NFT   (NFP/16)          // 157 face tiles
#define NTILE (HW/16)           // 1024 pixel tiles per image
#define FOCALF 12.0f
#define EPSF  1e-8f

typedef float v2f __attribute__((ext_vector_type(2)));
typedef float v8f __attribute__((ext_vector_type(8)));

// ---------------- pass 1: vertex transform (view z + ndc) ----------------
__global__ void k_transform(const float* __restrict__ verts,
                            float* __restrict__ zb, float* __restrict__ ndc) {
    int t = blockIdx.x * blockDim.x + threadIdx.x;
    if (t >= NB * NV) return;
    float x = verts[3*t+0], y = verts[3*t+1], z = verts[3*t+2];
    // view = verts @ diag(-1,1,-1) + (0,0,2)
    float vz = 2.0f - z;
    zb[t] = vz;
    ndc[2*t+0] = FOCALF * (-x) / vz;
    ndc[2*t+1] = FOCALF * ( y) / vz;
}

// -------- pass 2: per-face affine coefs written in WMMA-ready B layout ---
// Per face-tile ft: 32 lanes x 8 floats.
//   lane n      (n=0..15): [cx0,cy0, cx1,cy1, cx2,cy2, cxS,cyS]  (B rows K=0,1)
//   lane n+16            : [cc0,0,   cc1,0,   cc2,0,   ccS,0  ]  (B rows K=2,3)
// Degenerate (|area|<=EPS) and padded faces get all-zero coefs -> S=0<EPS -> never hit.
__global__ void k_face(const float* __restrict__ verts, const int* __restrict__ faces,
                       const float* __restrict__ zb, const float* __restrict__ ndc,
                       float* __restrict__ wmmaB, float* __restrict__ fn) {
    int t = blockIdx.x * blockDim.x + threadIdx.x;
    if (t >= NB * NFP) return;
    int b = t / NFP, f = t % NFP;
    int ft = f >> 4, n = f & 15;
    float* lo = wmmaB + ((size_t)(b*NFT + ft)*32 + n     ) * 8;  // lanes 0-15 slot
    float* hi = wmmaB + ((size_t)(b*NFT + ft)*32 + n + 16) * 8;  // lanes 16-31 slot

    float cx0=0,cy0=0,cc0=0, cx1=0,cy1=0,cc1=0, cx2=0,cy2=0,cc2=0, cxS=0,cyS=0,ccS=0;

    if (f < NF) {
        int i0 = faces[3*f+0], i1 = faces[3*f+1], i2 = faces[3*f+2];
        const float* nd = ndc + (size_t)b * NV * 2;
        const float* zz = zb  + (size_t)b * NV;
        float x0 = nd[2*i0], y0 = nd[2*i0+1];
        float x1 = nd[2*i1], y1 = nd[2*i1+1];
        float x2 = nd[2*i2], y2 = nd[2*i2+1];
        float z0 = zz[i0], z1 = zz[i1], z2 = zz[i2];

        float area = (x1-x0)*(y2-y0) - (y1-y0)*(x2-x0);
        if (fabsf(area) > EPSF) {
            float inv = 1.0f / area;
            // b0 = edge(v1,v2)/area : E(p) = (x2-x1)*(py-y1) - (y2-y1)*(px-x1)
            cx0 = -(y2-y1)*inv; cy0 = (x2-x1)*inv; cc0 = ((y2-y1)*x1 - (x2-x1)*y1)*inv;
            // b1 = edge(v2,v0)/area
            cx1 = -(y0-y2)*inv; cy1 = (x0-x2)*inv; cc1 = ((y0-y2)*x2 - (x0-x2)*y2)*inv;
            // b2 = 1 - b0 - b1
            cx2 = -cx0-cx1; cy2 = -cy0-cy1; cc2 = 1.0f-cc0-cc1;
            float iz0 = 1.0f/z0, iz1 = 1.0f/z1, iz2 = 1.0f/z2;
            cxS = cx0*iz0 + cx1*iz1 + cx2*iz2;
            cyS = cy0*iz0 + cy1*iz1 + cy2*iz2;
            ccS = cc0*iz0 + cc1*iz1 + cc2*iz2;
        }
        // world-space face normal (unnormalized cross product)
        const float* vb = verts + (size_t)b * NV * 3;
        float ax = vb[3*i0], ay = vb[3*i0+1], az = vb[3*i0+2];
        float e1x = vb[3*i1]-ax, e1y = vb[3*i1+1]-ay, e1z = vb[3*i1+2]-az;
        float e2x = vb[3*i2]-ax, e2y = vb[3*i2+1]-ay, e2z = vb[3*i2+2]-az;
        float* fno = fn + ((size_t)b*NF + f) * 3;
        fno[0] = e1y*e2z - e1z*e2y;
        fno[1] = e1z*e2x - e1x*e2z;
        fno[2] = e1x*e2y - e1y*e2x;
    }

    lo[0]=cx0; lo[1]=cy0; lo[2]=cx1; lo[3]=cy1; lo[4]=cx2; lo[5]=cy2; lo[6]=cxS; lo[7]=cyS;
    hi[0]=cc0; hi[1]=0.f; hi[2]=cc1; hi[3]=0.f; hi[4]=cc2; hi[5]=0.f; hi[6]=ccS; hi[7]=0.f;
}

// ------------- pass 3: vertex normals (deterministic gather) -------------
__global__ void k_vnorm(const int* __restrict__ faces, const float* __restrict__ fn,
                        float* __restrict__ vn) {
    int t = blockIdx.x * blockDim.x + threadIdx.x;
    if (t >= NB * NV) return;
    int b = t / NV, v = t % NV;
    const float* fnb = fn + (size_t)b * NF * 3;
    float ax = 0.f, ay = 0.f, az = 0.f;
    // three passes (corner 0, then 1, then 2) to mirror the reference's scatter order
    for (int k = 0; k < 3; ++k)
        for (int f = 0; f < NF; ++f)
            if (faces[3*f+k] == v) { ax += fnb[3*f]; ay += fnb[3*f+1]; az += fnb[3*f+2]; }
    float nl = sqrtf(ax*ax + ay*ay + az*az) + EPSF;
    vn[3*t+0] = ax/nl; vn[3*t+1] = ay/nl; vn[3*t+2] = az/nl;
}

// -------- pass 4: WMMA rasterizer — argmax(S) == argmin(depth) -----------
// Wave handles one 16-pixel tile. A (16x4 f32) = [px, py, 1, 0] per pixel row.
// B (4x16 f32) per affine function, pre-laid-out by k_face. D = A x B.
// depth = 1/S for inside & S>EPS; minimizing depth == maximizing S, so we
// track running max-S (initialized to EPS, folding in the S>EPS gate).
__global__ void __launch_bounds__(256) k_raster(const float* __restrict__ wmmaB,
                                                int* __restrict__ fidx) {
    int lane = threadIdx.x & 31;
    int wid  = blockIdx.x * (blockDim.x >> 5) + (threadIdx.x >> 5);
    int b    = wid >> 10;         // NTILE = 1024
    int tile = wid & (NTILE - 1);
    int l15  = lane & 15;
    int half = lane >> 4;

    // A-matrix (f32 16x4): lanes 0-15: v0=A[M][0]=px, v1=A[M][1]=py;
    //                      lanes 16-31: v0=A[M][2]=1, v1=A[M][3]=0
    int pix  = tile * 16 + l15;
    int prow = pix >> 7, pcol = pix & 127;
    float px = 1.0f - (2.0f*pcol + 1.0f) * (1.0f/IMG);
    float py = 1.0f - (2.0f*prow + 1.0f) * (1.0f/IMG);
    v2f A;
    A.x = half ? 1.0f : px;
    A.y = half ? 0.0f : py;

    float smax[8]; int fsel[8];
    #pragma unroll
    for (int r = 0; r < 8; ++r) { smax[r] = EPSF; fsel[r] = 0; }

    const float* bbase = wmmaB + (size_t)b * NFT * 32 * 8;
    for (int ft = 0; ft < NFT; ++ft) {
        // each lane loads exactly its own B-operand data: 32B, two aligned b128s
        const float* p = bbase + ((size_t)ft*32 + lane) * 8;
        float4 q0 = *(const float4*)(p + 0);
        float4 q1 = *(const float4*)(p + 4);
        v2f B0; B0.x = q0.x; B0.y = q0.y;
        v2f B1; B1.x = q0.z; B1.y = q0.w;
        v2f B2; B2.x = q1.x; B2.y = q1.y;
        v2f BS; BS.x = q1.z; BS.y = q1.w;

        v8f zc = {};
        v8f d0 = __builtin_amdgcn_wmma_f32_16x16x4_f32(false, A, false, B0, (short)0, zc, false, false);
        v8f d1 = __builtin_amdgcn_wmma_f32_16x16x4_f32(false, A, false, B1, (short)0, zc, false, false);
        v8f d2 = __builtin_amdgcn_wmma_f32_16x16x4_f32(false, A, false, B2, (short)0, zc, false, false);
        v8f dS = __builtin_amdgcn_wmma_f32_16x16x4_f32(false, A, false, BS, (short)0, zc, false, false);

        int face = ft*16 + l15;      // this lane's face column
        #pragma unroll
        for (int r = 0; r < 8; ++r) {
            float mn = fminf(fminf(d0[r], d1[r]), d2[r]);   // inside <=> mn >= 0
            float S  = dS[r];
            bool take = (mn >= 0.0f) && (S > smax[r]);      // strict > keeps earliest face
            if (take) { smax[r] = S; fsel[r] = face; }
        }
    }

    // reduce over the 16 face columns in each half-wave; ties -> smaller face index
    #pragma unroll
    for (int r = 0; r < 8; ++r) {
        #pragma unroll
        for (int m = 1; m < 16; m <<= 1) {
            float os = __shfl_xor(smax[r], m, 32);
            int   oi = __shfl_xor(fsel[r], m, 32);
            if (os > smax[r] || (os == smax[r] && oi < fsel[r])) { smax[r] = os; fsel[r] = oi; }
        }
    }
    if (l15 == 0) {
        #pragma unroll
        for (int r = 0; r < 8; ++r) {
            int p = tile*16 + r + 8*half;   // lanes 0-15 -> rows 0-7, lanes 16-31 -> rows 8-15
            fidx[(size_t)b*HW + p] = (smax[r] > EPSF) ? fsel[r] : -1;
        }
    }
}

// ----------------------- pass 5: Phong shading ---------------------------
__global__ void k_shade(const float* __restrict__ verts, const int* __restrict__ faces,
                        const float* __restrict__ zb, const float* __restrict__ ndc,
                        const float* __restrict__ vn, const int* __restrict__ fidx,
                        float* __restrict__ out) {
    int t = blockIdx.x * blockDim.x + threadIdx.x;
    if (t >= NB * HW) return;
    int b = t / HW, p = t % HW;
    float* img   = out + (size_t)b * 3 * HW;
    float* alpha = out + (size_t)NB * 3 * HW + (size_t)b * HW;

    int fi = fidx[t];
    if (fi < 0) {
        img[0*HW+p] = 255.0f; img[1*HW+p] = 255.0f; img[2*HW+p] = 255.0f;
        alpha[p] = 0.0f;
        return;
    }
    int prow = p >> 7, pcol = p & 127;
    float px = 1.0f - (2.0f*pcol + 1.0f) * (1.0f/IMG);
    float py = 1.0f - (2.0f*prow + 1.0f) * (1.0f/IMG);

    int i0 = faces[3*fi+0], i1 = faces[3*fi+1], i2 = faces[3*fi+2];
    const float* nd = ndc + (size_t)b * NV * 2;
    const float* zz = zb  + (size_t)b * NV;
    float x0 = nd[2*i0], y0 = nd[2*i0+1];
    float x1 = nd[2*i1], y1 = nd[2*i1+1];
    float x2 = nd[2*i2], y2 = nd[2*i2+1];
    float z0 = zz[i0], z1 = zz[i1], z2 = zz[i2];

    float ar = (x1-x0)*(y2-y0) - (y1-y0)*(x2-x0);
    if (fabsf(ar) < EPSF) ar = EPSF;
    float bb0 = ((x2-x1)*(py-y1) - (y2-y1)*(px-x1)) / ar;
    float bb1 = ((x0-x2)*(py-y2) - (y0-y2)*(px-x2)) / ar;
    float bb2 = 1.0f - bb0 - bb1;
    float w0 = bb0/z0, w1 = bb1/z1, w2 = bb2/z2;
    float sw = w0 + w1 + w2 + EPSF;
    float pc0 = w0/sw, pc1 = w1/sw, pc2 = w2/sw;

    const float* vb  = verts + (size_t)b * NV * 3;
    const float* vnb = vn    + (size_t)b * NV * 3;
    float posx = pc0*vb[3*i0+0] + pc1*vb[3*i1+0] + pc2*vb[3*i2+0];
    float posy = pc0*vb[3*i0+1] + pc1*vb[3*i1+1] + pc2*vb[3*i2+1];
    float posz = pc0*vb[3*i0+2] + pc1*vb[3*i1+2] + pc2*vb[3*i2+2];
    float nx = pc0*vnb[3*i0+0] + pc1*vnb[3*i1+0] + pc2*vnb[3*i2+0];
    float ny = pc0*vnb[3*i0+1] + pc1*vnb[3*i1+1] + pc2*vnb[3*i2+1];
    float nz = pc0*vnb[3*i0+2] + pc1*vnb[3*i1+2] + pc2*vnb[3*i2+2];
    float nl = sqrtf(nx*nx + ny*ny + nz*nz) + EPSF;
    nx /= nl; ny /= nl; nz /= nl;

    float lx = 0.0f - posx, ly = 1.0f - posy, lz = 3.0f - posz;      // LIGHT_LOC
    float ll = sqrtf(lx*lx + ly*ly + lz*lz) + EPSF;
    lx /= ll; ly /= ll; lz /= ll;
    float vx = -posx, vy = -posy, vz_ = 2.0f - posz;                 // CAM_POS
    float vl = sqrtf(vx*vx + vy*vy + vz_*vz_) + EPSF;
    vx /= vl; vy /= vl; vz_ /= vl;

    float ndl_raw = nx*lx + ny*ly + nz*lz;
    float ndl = fmaxf(ndl_raw, 0.0f);
    float rx = 2.0f*ndl_raw*nx - lx;
    float ry = 2.0f*ndl_raw*ny - ly;
    float rz = 2.0f*ndl_raw*nz - lz;
    float sc = fmaxf(rx*vx + ry*vy + rz*vz_, 0.0f);
    float s2 = sc*sc, s4 = s2*s2, s8 = s4*s4, s10 = s8*s2;           // sc^10

    float amb_diff = 0.5f + 0.3f*ndl;                                // AMB + DIFF*ndl
    float spec = 0.12f * s10;                                        // SPEC_L*MAT_SPEC
    img[0*HW+p] = ((142.0f/255.0f)*amb_diff + spec) * 255.0f;
    img[1*HW+p] = ((179.0f/255.0f)*amb_diff + spec) * 255.0f;
    img[2*HW+p] = ((247.0f/255.0f)*amb_diff + spec) * 255.0f;
    alpha[p] = 1.0f;
}

// --------------------------------------------------------------------------
extern "C" void kernel_launch(void* const* d_in, const int* in_sizes, int n_in,
                              void* d_out, int out_size, void* d_ws, size_t ws_size,
                              hipStream_t stream) {
    const float* verts = (const float*)d_in[0];   // [2,1300,3] f32
    const int*   faces = (const int*)d_in[1];     // [2500,3] i32
    float* out = (float*)d_out;                   // images [2,3,128,128] ++ alphas [2,1,128,128]

    float* ws    = (float*)d_ws;
    float* zbuf  = ws;                                  // NB*NV
    float* ndc   = zbuf + NB*NV;                        // NB*NV*2
    float* wmmaB = ndc  + NB*NV*2;                      // NB*NFT*32*8
    float* fn    = wmmaB + (size_t)NB*NFT*32*8;         // NB*NF*3
    float* vn    = fn   + (size_t)NB*NF*3;              // NB*NV*3
    int*   fidx  = (int*)(vn + (size_t)NB*NV*3);        // NB*HW ints
    // total: ~144K floats (~576 KB) of workspace

    k_transform<<<(NB*NV  + 255)/256, 256, 0, stream>>>(verts, zbuf, ndc);
    k_face     <<<(NB*NFP + 255)/256, 256, 0, stream>>>(verts, faces, zbuf, ndc, wmmaB, fn);
    k_vnorm    <<<(NB*NV  + 255)/256, 256, 0, stream>>>(faces, fn, vn);
    k_raster   <<<(NB*NTILE)/8, 256, 0, stream>>>(wmmaB, fidx);   // 8 waves/block, 1 tile/wave
    k_shade    <<<(NB*HW  + 255)/256, 256, 0, stream>>>(verts, faces, zbuf, ndc, vn, fidx, out);
}